// GaussianSplattingModel_17454747091438
// MI455X (gfx1250) — compile-verified
//
#include <hip/hip_runtime.h>
#include <hip/hip_bf16.h>
#include <math.h>

// CDNA5 / gfx1250 Gaussian splatting.
//  - Rasterizer: V_WMMA_F32_16X16X4_F32 evaluates the Gaussian quadratic form
//    for 16-gaussian x 16-pixel tiles (two tiles per chunk -> 32 px per wave).
//  - Sorted per-gaussian block (52KB) staged into LDS once per workgroup via
//    the Tensor Data Mover (tensor_load_to_lds + s_wait_tensorcnt).

typedef __attribute__((ext_vector_type(2))) float v2f;
typedef __attribute__((ext_vector_type(8))) float v8f;
typedef __attribute__((ext_vector_type(4))) unsigned int u32x4;
typedef __attribute__((ext_vector_type(4))) int i32x4;
typedef __attribute__((ext_vector_type(8))) int i32x8;

#define NG   1024
#define IMH  128
#define IMW  128

// sorted block: par8 (8192 f) | colop (4096 f) | bbox (1024 u32) = 13312 dwords
#define SBLK_DW 13312

// ---------------------------------------------------------------------------
// Kernel 1: per-gaussian preprocessing.
// ---------------------------------------------------------------------------
__global__ void gs_preprocess(const float* __restrict__ cam,
                              const float* __restrict__ means,
                              const float* __restrict__ scales,
                              const float* __restrict__ rots,
                              const float* __restrict__ colors,
                              const float* __restrict__ opac,
                              const int*   __restrict__ focal,
                              float* __restrict__ depth,
                              float* __restrict__ par8,
                              float4* __restrict__ colop,
                              unsigned int* __restrict__ bbox)
{
    int i = blockIdx.x * blockDim.x + threadIdx.x;
    if (i >= NG) return;

    float f = (float)focal[0];

    float Rc[3][3], tc[3];
#pragma unroll
    for (int r = 0; r < 3; ++r) {
        Rc[r][0] = cam[r*4+0]; Rc[r][1] = cam[r*4+1]; Rc[r][2] = cam[r*4+2];
        tc[r]    = cam[r*4+3];
    }

    float mn[3] = { means[i*3+0], means[i*3+1], means[i*3+2] };
    float mc[3];
#pragma unroll
    for (int r = 0; r < 3; ++r)
        mc[r] = Rc[r][0]*mn[0] + Rc[r][1]*mn[1] + Rc[r][2]*mn[2] + tc[r];

    float dep = mc[2];
    float Z   = fmaxf(dep, 1e-10f);
    float cx  = f * mc[0] / Z + 0.5f * (float)IMW;
    float cy  = f * mc[1] / Z + 0.5f * (float)IMH;

    float qw = rots[i*4+0], qx = rots[i*4+1], qy = rots[i*4+2], qz = rots[i*4+3];
    float inv = 1.0f / sqrtf(qw*qw + qx*qx + qy*qy + qz*qz);
    qw *= inv; qx *= inv; qy *= inv; qz *= inv;
    float G[3][3] = {
        {1.f-2.f*(qy*qy+qz*qz), 2.f*(qx*qy-qw*qz),     2.f*(qx*qz+qw*qy)},
        {2.f*(qx*qy+qw*qz),     1.f-2.f*(qx*qx+qz*qz), 2.f*(qy*qz-qw*qx)},
        {2.f*(qx*qz-qw*qy),     2.f*(qy*qz+qw*qx),     1.f-2.f*(qx*qx+qy*qy)}};

    float s0 = scales[i*3+0], s1 = scales[i*3+1], s2s = scales[i*3+2];
    float s2[3] = { s0*s0, s1*s1, s2s*s2s };

    float C3[3][3];
#pragma unroll
    for (int j = 0; j < 3; ++j)
#pragma unroll
        for (int k = 0; k < 3; ++k)
            C3[j][k] = G[j][0]*s2[0]*G[k][0] + G[j][1]*s2[1]*G[k][1] + G[j][2]*s2[2]*G[k][2];

    float T[3][3];
#pragma unroll
    for (int j = 0; j < 3; ++j)
#pragma unroll
        for (int k = 0; k < 3; ++k)
            T[j][k] = Rc[j][0]*C3[0][k] + Rc[j][1]*C3[1][k] + Rc[j][2]*C3[2][k];
    float CC[3][3];
#pragma unroll
    for (int j = 0; j < 3; ++j)
#pragma unroll
        for (int l = 0; l < 3; ++l)
            CC[j][l] = T[j][0]*Rc[l][0] + T[j][1]*Rc[l][1] + T[j][2]*Rc[l][2];

    float fz = f / Z;
    float jx = -f * mc[0] / (Z * Z);
    float jy = -f * mc[1] / (Z * Z);
    float J0[3] = { fz, 0.f, jx };
    float J1[3] = { 0.f, fz, jy };

    float t0[3], t1[3];
#pragma unroll
    for (int p = 0; p < 3; ++p) {
        t0[p] = CC[p][0]*J0[0] + CC[p][1]*J0[1] + CC[p][2]*J0[2];
        t1[p] = CC[p][0]*J1[0] + CC[p][1]*J1[1] + CC[p][2]*J1[2];
    }
    float a  = J0[0]*t0[0] + J0[1]*t0[1] + J0[2]*t0[2] + 1e-6f;
    float b  = J0[0]*t1[0] + J0[1]*t1[1] + J0[2]*t1[2];
    float c_ = J1[0]*t0[0] + J1[1]*t0[1] + J1[2]*t0[2];
    float dd = J1[0]*t1[0] + J1[1]*t1[1] + J1[2]*t1[2] + 1e-6f;

    float det = a*dd - b*c_;
    float iA =  dd / det;
    float iB = -b  / det;
    float iC = -c_ / det;
    float iD =  a  / det;

    float radius = 3.0f * fmaxf(sqrtf(a + 1e-10f), sqrtf(dd + 1e-10f));

    bool valid  = dep > 0.f;
    bool culled = (cx < -0.5f*(float)IMW) || (cx > 1.5f*(float)IMW) ||
                  (cy < -0.5f*(float)IMH) || (cy > 1.5f*(float)IMH);
    bool act = valid && !culled;

    float op  = opac[i];
    float opa = act ? (1.0f / (1.0f + __expf(-op))) : 0.0f;

    float x0 = fmaxf(0.f, truncf(cx - radius));
    float x1 = fminf((float)IMW, truncf(cx + radius + 1.f));
    float y0 = fmaxf(0.f, truncf(cy - radius));
    float y1 = fminf((float)IMH, truncf(cy + radius + 1.f));

    unsigned int bb = 0u;
    if (act) {
        unsigned int ix0 = (unsigned int)(int)fminf(fmaxf(x0, 0.f), 255.f);
        unsigned int ix1 = (unsigned int)(int)fminf(fmaxf(x1, 0.f), 255.f);
        unsigned int iy0 = (unsigned int)(int)fminf(fmaxf(y0, 0.f), 255.f);
        unsigned int iy1 = (unsigned int)(int)fminf(fmaxf(y1, 0.f), 255.f);
        bb = ix0 | (ix1 << 8) | (iy0 << 16) | (iy1 << 24);
    }

    depth[i] = dep;
    par8[i*8+0] = iA;
    par8[i*8+1] = iD;
    par8[i*8+2] = iB + iC;
    par8[i*8+3] = cx;
    par8[i*8+4] = cy;
    par8[i*8+5] = 0.f;
    par8[i*8+6] = 0.f;
    par8[i*8+7] = 0.f;
    colop[i] = make_float4(colors[i*3+0], colors[i*3+1], colors[i*3+2], opa);
    bbox[i]  = bb;
}

// ---------------------------------------------------------------------------
// Kernel 2: stable descending-depth rank (O(N^2)) + scatter to sorted arrays.
// ---------------------------------------------------------------------------
__global__ void gs_rank_scatter(const float* __restrict__ depth,
                                const float* __restrict__ par8,
                                const float4* __restrict__ colop,
                                const unsigned int* __restrict__ bbox,
                                float* __restrict__ spar8,
                                float4* __restrict__ scolop,
                                unsigned int* __restrict__ sbbox)
{
    int i = blockIdx.x * blockDim.x + threadIdx.x;
    if (i >= NG) return;
    float di = depth[i];
    int r = 0;
    for (int j = 0; j < NG; ++j) {
        float dj = depth[j];
        r += ((dj > di) || (dj == di && j < i)) ? 1 : 0;
    }
#pragma unroll
    for (int k = 0; k < 8; ++k) spar8[r*8+k] = par8[i*8+k];
    scolop[r] = colop[i];
    sbbox[r]  = bbox[i];
}

// ---------------------------------------------------------------------------
// Kernel 3: rasterizer. One wave32 per 32-pixel row span; two WMMA tiles per
// 16-gaussian chunk (shared A matrix, single anchor). Sorted gaussian block
// staged to LDS via the Tensor Data Mover once per workgroup.
// ---------------------------------------------------------------------------
__global__ void __launch_bounds__(256)
gs_raster(const float* __restrict__ sorted_blk,   // SBLK_DW dwords, contiguous
          float* __restrict__ out)
{
    __shared__ __align__(16) float smem[SBLK_DW];

    // --- Stage sorted block into LDS with the TDM (one DMA per workgroup) ---
    if (threadIdx.x == 0) {
        unsigned long long ga = (unsigned long long)(uintptr_t)sorted_blk;
        unsigned int lds_off  = (unsigned int)(unsigned long long)(uintptr_t)&smem[0];

        u32x4 g0;
        g0.x = 1u;                                    // count=1, user mode
        g0.y = lds_off;                               // lds_addr (bytes)
        g0.z = (unsigned int)(ga & 0xffffffffull);    // global_addr[31:0]
        g0.w = (unsigned int)((ga >> 32) & 0x01ffffffull) | (2u << 30); // type=2

        i32x8 g1;
        g1[0] = (2 << 16);                 // data_size = 4 bytes; mask/flags 0
        g1[1] = (int)(((unsigned)SBLK_DW & 0xffffu) << 16); // tensor_dim0 lo16
        g1[2] = (int)((((unsigned)SBLK_DW >> 16) & 0xffffu) |
                      (1u << 16));         // tensor_dim0 hi16 | tensor_dim1=1
        g1[3] = (int)(((unsigned)SBLK_DW & 0xffffu) << 16); // tile_dim0
        g1[4] = 1;                         // tile_dim1=1, tile_dim2=0
        g1[5] = SBLK_DW;                   // tensor_dim0_stride lo32
        g1[6] = 0;                         // stride hi16 | dim1_stride lo16
        g1[7] = 0;

        i32x4 g2 = {0, 0, 0, 0};
        i32x4 g3 = {0, 0, 0, 0};
#if defined(__clang_major__) && (__clang_major__ >= 23)
        i32x8 g4 = {0, 0, 0, 0, 0, 0, 0, 0};
        __builtin_amdgcn_tensor_load_to_lds(g0, g1, g2, g3, g4, 0);
#else
        __builtin_amdgcn_tensor_load_to_lds(g0, g1, g2, g3, 0);
#endif
        __builtin_amdgcn_s_wait_tensorcnt(0);
    }
    __syncthreads();

    const float4*       P4 = (const float4*)smem;            // 2 float4 / gauss
    const float4*       CO = (const float4*)(smem + 8192);   // (r,g,b,opa)
    const unsigned int* BB = (const unsigned int*)(smem + 12288);

    int tid  = threadIdx.x;
    int lane = tid & 31;
    int wid  = blockIdx.x * 8 + (tid >> 5);   // 512 waves total
    int py   = wid >> 2;                      // row 0..127
    int px0  = (wid & 3) << 5;                // span start 0,32,64,96

    int   m  = lane & 15;                     // gaussian slot / pixel-in-tile
    int   h  = lane >> 4;                     // K-half / tile-half selector
    float ax = (float)px0;
    float ay = (float)py;

    // B matrices (4x16 pixel features), compile-time per lane.
    //   tile A: u = m        (pixels px0 .. px0+15)
    //   tile B: u = m + 16   (pixels px0+16 .. px0+31)
    float uA = (float)m;
    float uB = (float)(m + 16);
    v2f bA, bB;
    bA.x = h ? 1.0f : uA * uA;   bA.y = h ? 0.0f : uA;
    bB.x = h ? 1.0f : uB * uB;   bB.y = h ? 0.0f : uB;

    int px = px0 + lane;  // this lane's composited pixel x (all 32 lanes useful)

    float alpha = 0.f, cr = 0.f, cg = 0.f, cb = 0.f;

    for (int chunk = 0; chunk < NG / 16; ++chunk) {
        int s = chunk * 16 + m;
        float4 p0 = P4[s*2 + 0];     // (ga, gd, Bc, cx)
        float4 p1 = P4[s*2 + 1];     // (cy, 0, 0, 0)
        float ga = p0.x, gd = p0.y, Bc = p0.z;
        float ccx = p0.w - ax;
        float ccy = p1.x - ay;

        // A (16x4), row v=0 baked in: Q(u) = ga*u^2 + f1*u + f2
        float f1 = -(2.f*ga*ccx + Bc*ccy);
        float f2 = ga*ccx*ccx + Bc*ccx*ccy + gd*ccy*ccy;
        v2f amat;
        amat.x = h ? f2 : ga;
        amat.y = h ? 0.f : f1;

        v8f qA = {};
        qA = __builtin_amdgcn_wmma_f32_16x16x4_f32(
                false, amat, false, bA, (short)0, qA, false, false);
        v8f qB = {};
        qB = __builtin_amdgcn_wmma_f32_16x16x4_f32(
                false, amat, false, bB, (short)0, qB, false, false);

        // D layout: lane L<16 -> (M=j, N=L); lane L>=16 -> (M=j+8, N=L-16).
        // Low lanes composite tile-A pixels, high lanes tile-B pixels.
        float qg[16];
#pragma unroll
        for (int j = 0; j < 8; ++j) {
            float oqA = __shfl(qA[j], m | 16, 32);   // g(j+8) for low lanes
            float oqB = __shfl(qB[j], m,      32);   // g(j)   for high lanes
            qg[j]     = h ? oqB   : qA[j];
            qg[j + 8] = h ? qB[j] : oqA;
        }

        int gbase = chunk * 16;
#pragma unroll
        for (int j = 0; j < 16; ++j) {
            float4 co       = CO[gbase + j];
            unsigned int bb = BB[gbase + j];
            int x0 = (int)(bb & 255u);
            int x1 = (int)((bb >> 8)  & 255u);
            int y0 = (int)((bb >> 16) & 255u);
            int y1 = (int)((bb >> 24) & 255u);
            bool inbox = (px >= x0) && (px < x1) && (py >= y0) && (py < y1);
            float w  = inbox ? __expf(-0.5f * qg[j]) * co.w : 0.f;
            float wc = w * (1.f - alpha);
            alpha += wc;
            cr += co.x * wc;
            cg += co.y * wc;
            cb += co.z * wc;
        }
    }

    int o = (py * IMW + px) * 3;
    out[o + 0] = cr;
    out[o + 1] = cg;
    out[o + 2] = cb;
}

// ---------------------------------------------------------------------------
extern "C" void kernel_launch(void* const* d_in, const int* in_sizes, int n_in,
                              void* d_out, int out_size, void* d_ws, size_t ws_size,
                              hipStream_t stream)
{
    (void)in_sizes; (void)n_in; (void)out_size; (void)ws_size;

    const float* cam    = (const float*)d_in[0];
    const float* means  = (const float*)d_in[1];
    const float* scales = (const float*)d_in[2];
    const float* rots   = (const float*)d_in[3];
    const float* colors = (const float*)d_in[4];
    const float* opac   = (const float*)d_in[5];
    // d_in[6]=H, d_in[7]=W (fixed 128), d_in[8]=focal
    const int* focal    = (const int*)d_in[8];

    float* ws = (float*)d_ws;
    float*        depth  = ws;                          // 1024 f
    float*        par8   = ws + 1024;                   // 8192 f
    float4*       colop  = (float4*)(ws + 9216);        // 4096 f
    unsigned int* bbox   = (unsigned int*)(ws + 13312); // 1024 u32
    // sorted block, contiguous: par8 | colop | bbox  (SBLK_DW dwords)
    float*        spar8  = ws + 14336;                  // 8192 f
    float4*       scolop = (float4*)(ws + 22528);       // 4096 f
    unsigned int* sbbox  = (unsigned int*)(ws + 26624); // 1024 u32

    float* out = (float*)d_out;

    gs_preprocess<<<4, 256, 0, stream>>>(cam, means, scales, rots, colors, opac,
                                         focal, depth, par8, colop, bbox);
    gs_rank_scatter<<<4, 256, 0, stream>>>(depth, par8, colop, bbox,
                                           spar8, scolop, sbbox);
    gs_raster<<<64, 256, 0, stream>>>(spar8, out);
}